// FullyConnected_60730837565842
// MI455X (gfx1250) — compile-verified
//
#include <hip/hip_runtime.h>

// Problem constants (from setup_inputs): x[256,1024], weight[512,1024], bias[512]
#define TOKENS 256
#define KDIM   1024
#define NDIM   512

// Tiling
#define BT  32            // token rows per block
#define NT  64            // output cols per block
#define KC  64            // K chunk staged in LDS
#define NKC (KDIM / KC)   // 16 chunks

// Padded LDS strides (floats). Rows stay 16B aligned; padding kills bank conflicts.
#define XS_STRIDE 36      // 36*4 = 144B row, 144 % 16 == 0
#define WS_STRIDE 68      // 68*4 = 272B row, 272 % 16 == 0

#if defined(__has_builtin)
#if __has_builtin(__builtin_amdgcn_global_load_async_to_lds_b32) && \
    __has_builtin(__builtin_amdgcn_s_wait_asynccnt)
#define USE_ASYNC_LDS 1
#endif
#endif

// One 4-byte global->LDS copy. Async path uses the CDNA5 DMA (ASYNCcnt);
// per-lane independent global & LDS addresses let us transpose in flight.
__device__ __forceinline__ void copy_g2l_b32(const float* g, float* l) {
#if defined(USE_ASYNC_LDS)
  __builtin_amdgcn_global_load_async_to_lds_b32(
      (__attribute__((address_space(1))) int*)(g),
      (__attribute__((address_space(3))) int*)(l),
      /*offset=*/0, /*cpol=*/0);
#else
  *l = *g;   // synchronous fallback; __syncthreads() fences it
#endif
}

__device__ __forceinline__ void wait_async_all() {
#if defined(USE_ASYNC_LDS)
  __builtin_amdgcn_s_wait_asynccnt(0);
#endif
}

__global__ __launch_bounds__(256) void
mam_fc_kernel(const float* __restrict__ x,      // [TOKENS][KDIM]
              const float* __restrict__ wgt,    // [NDIM][KDIM]  (torch layout; w[k][n] = wgt[n][k])
              const float* __restrict__ bias,   // [NDIM]
              float* __restrict__ out)          // [TOKENS][NDIM]
{
  // Double-buffered LDS tiles, stored K-major so the inner loop reads are
  // contiguous in b (xs) and n (ws).
  __shared__ __align__(16) float xs[2][KC * XS_STRIDE];  // 2 * 9216 B
  __shared__ __align__(16) float ws[2][KC * WS_STRIDE];  // 2 * 17408 B

  const int tid = (int)threadIdx.x;
  const int tx  = tid & 31;   // n group: n = n0 + 2*tx + {0,1}
  const int ty  = tid >> 5;   // b group: b = b0 + 4*ty + {0..3}
  const int n0  = (int)blockIdx.x * NT;
  const int b0  = (int)blockIdx.y * BT;

  // Stage one K-chunk into LDS buffer `buf`, transposing to [k][b] / [k][n].
  // e = tid + 256*i keeps each wave's 32 lanes on 32 consecutive k values of
  // one row -> fully coalesced 128B global reads.
  auto load_tile = [&](int buf, int kc) {
    const int k0 = kc * KC;
#pragma unroll
    for (int i = 0; i < (BT * KC) / 256; ++i) {          // 8 copies/thread
      const int e = tid + 256 * i;
      const int k = e & (KC - 1);
      const int b = e >> 6;
      copy_g2l_b32(x + (size_t)(b0 + b) * KDIM + (k0 + k),
                   &xs[buf][k * XS_STRIDE + b]);
    }
#pragma unroll
    for (int i = 0; i < (NT * KC) / 256; ++i) {          // 16 copies/thread
      const int e = tid + 256 * i;
      const int k = e & (KC - 1);
      const int n = e >> 6;
      copy_g2l_b32(wgt + (size_t)(n0 + n) * KDIM + (k0 + k),
                   &ws[buf][k * WS_STRIDE + n]);
    }
  };

  // 4x2 micro-tile of online max/min accumulators.
  float mx[4][2], mn[4][2];
#pragma unroll
  for (int i = 0; i < 4; ++i)
#pragma unroll
    for (int j = 0; j < 2; ++j) {
      mx[i][j] = -__builtin_inff();
      mn[i][j] =  __builtin_inff();
    }

  load_tile(0, 0);

  for (int kc = 0; kc < NKC; ++kc) {
    wait_async_all();     // this wave's DMA for chunk kc has landed in LDS
    __syncthreads();      // whole tile present; previous compute buffer free
    if (kc + 1 < NKC)
      load_tile((kc + 1) & 1, kc + 1);   // DMA overlaps compute below

    const float* xrow = &xs[kc & 1][4 * ty];   // 16B aligned (ty*16B + k*144B)
    const float* wrow = &ws[kc & 1][2 * tx];   // 8B aligned  (tx*8B + k*272B)

#pragma unroll 8
    for (int k = 0; k < KC; ++k) {
      const float4 xv = *(const float4*)(xrow + (size_t)k * XS_STRIDE); // ds_load_b128
      const float2 wv = *(const float2*)(wrow + (size_t)k * WS_STRIDE); // ds_load_b64
      const float xa[4] = {xv.x, xv.y, xv.z, xv.w};
      const float wa[2] = {wv.x, wv.y};
#pragma unroll
      for (int i = 0; i < 4; ++i)
#pragma unroll
        for (int j = 0; j < 2; ++j) {
          const float p = xa[i] * wa[j];
          mx[i][j] = fmaxf(mx[i][j], p);
          mn[i][j] = fminf(mn[i][j], p);
        }
    }
  }

  // Epilogue: out = max + min + bias, 8B vector stores (row-contiguous pairs).
  const float2 bv = *(const float2*)(bias + n0 + 2 * tx);
#pragma unroll
  for (int i = 0; i < 4; ++i) {
    float2 o;
    o.x = mx[i][0] + mn[i][0] + bv.x;
    o.y = mx[i][1] + mn[i][1] + bv.y;
    *(float2*)(out + (size_t)(b0 + 4 * ty + i) * NDIM + (n0 + 2 * tx)) = o;
  }
}

extern "C" void kernel_launch(void* const* d_in, const int* in_sizes, int n_in,
                              void* d_out, int out_size, void* d_ws, size_t ws_size,
                              hipStream_t stream) {
  (void)in_sizes; (void)n_in; (void)out_size; (void)d_ws; (void)ws_size;
  const float* x    = (const float*)d_in[0];   // [256,1024]
  const float* wgt  = (const float*)d_in[1];   // [512,1024]
  const float* bias = (const float*)d_in[2];   // [512]
  float*       out  = (float*)d_out;           // [256,512]

  dim3 grid(NDIM / NT, TOKENS / BT);           // (8, 8) = 64 blocks
  mam_fc_kernel<<<grid, 256, 0, stream>>>(x, wgt, bias, out);
}